// C_Attention_24781961298239
// MI455X (gfx1250) — compile-verified
//
#include <hip/hip_runtime.h>

typedef float        v2f  __attribute__((ext_vector_type(2)));
typedef float        v8f  __attribute__((ext_vector_type(8)));
typedef unsigned int v4u  __attribute__((ext_vector_type(4)));
typedef int          v4i  __attribute__((ext_vector_type(4)));
typedef int          v8i  __attribute__((ext_vector_type(8)));

#define B_   16
#define T_   1024
#define N_   1024
#define D_   256
#define WIN_ 64

// ---------------------------------------------------------------------------
// Kernel 1: zero the alignments region (B*N*T floats = 64 MB). Fully coalesced
// float4 stores; 4096 blocks x 256 threads x 4 stores covers it exactly.
// ---------------------------------------------------------------------------
__global__ __launch_bounds__(256) void zero_align_f4(float4* __restrict__ p) {
    const size_t tid    = (size_t)blockIdx.x * 256 + threadIdx.x;
    const size_t stride = (size_t)4096 * 256;
    const float4 z = make_float4(0.f, 0.f, 0.f, 0.f);
    p[tid]              = z;
    p[tid + stride]     = z;
    p[tid + 2 * stride] = z;
    p[tid + 3 * stride] = z;
}

// ---------------------------------------------------------------------------
// TDM: DMA the 64x256 fp32 V window (64 KB) into LDS via tensor_load_to_lds.
// D# built per CDNA5 ISA (2D tile, data_size=4B).
// Tracked with TENSORcnt; single wave per workgroup waits before consuming.
// ---------------------------------------------------------------------------
__device__ static inline void tdm_load_vwin(const float* gsrc, unsigned int lds_byte_off) {
    const unsigned long long ga = (unsigned long long)(uintptr_t)gsrc;

    v4u g0;
    g0[0] = 1u;                                            // count=1, user mode, no gather
    g0[1] = lds_byte_off;                                  // lds_addr (bytes)
    g0[2] = (unsigned int)ga;                              // global_addr[31:0]
    g0[3] = (unsigned int)((ga >> 32) & 0x01FFFFFFu)       // global_addr[56:32]
          | (2u << 30);                                    // type = 2 ("image")

    v8i g1;
    g1[0] = (int)(2u << 16);          // wg_mask=0 | data_size=2 (4 bytes) | no pad/iter
    g1[1] = (int)((unsigned)D_ << 16);    // tensor_dim0[15:0] = 256   (bits 63:48)
    g1[2] = (int)((unsigned)WIN_ << 16);  // dim0 hi=0 | tensor_dim1[15:0] = 64
    g1[3] = (int)((unsigned)D_ << 16);    // dim1 hi=0 | tile_dim0 = 256
    g1[4] = WIN_;                     // tile_dim1 = 64 | tile_dim2 = 0
    g1[5] = D_;                       // tensor_dim0_stride[31:0] = 256 elements
    g1[6] = 0;                        // stride0 hi | tensor_dim1_stride lo (unused, 2D)
    g1[7] = 0;

    const v4i z4 = {0, 0, 0, 0};
#if defined(__clang_major__) && (__clang_major__ <= 22)
    __builtin_amdgcn_tensor_load_to_lds(g0, g1, z4, z4, 0);
#else
    const v8i z8 = {0, 0, 0, 0, 0, 0, 0, 0};
    __builtin_amdgcn_tensor_load_to_lds(g0, g1, z4, z4, z8, 0);
#endif
}

// ---------------------------------------------------------------------------
// Kernel 2: windowed attention, one wave32 per (batch b, 16-row Q tile).
//   TDM V window -> LDS  (overlapped with QK phase, TENSORcnt)
//   S = Q_tile(16xD) @ Kwin^T(Dx64)  via v_wmma_f32_16x16x4_f32
//   softmax over 64 (mask outside window == exact 0 after exp)
//   O = P(16x64) @ Vlds(64xD)        via v_wmma_f32_16x16x4_f32
// ---------------------------------------------------------------------------
__global__ __launch_bounds__(32) void attn_win_wmma(
    const float* __restrict__ Q, const float* __restrict__ K,
    const float* __restrict__ V, const int* __restrict__ prevArr,
    float* __restrict__ R, float* __restrict__ align_out,
    float* __restrict__ maxatt)
{
    __shared__ __align__(16) float Vlds[WIN_ * D_];   // 64 KB V stage (TDM target)
    __shared__ float P[16 * 65];                      // probs, stride 65 (bank-safe)

    const int lane = threadIdx.x;         // 0..31 (wave32)
    const int lrow = lane & 15;           // row (A frag) or col (B/C frag) index
    const int koff = (lane >> 4) * 2;     // lanes 16..31 hold k+2,k+3 per ISA layout
    const int hi8  = (lane >> 4) * 8;     // C/D frag: lanes 16..31 hold rows m+8

    const int blk  = blockIdx.x;
    const int b    = blk >> 6;            // 64 tiles per batch (T/16)
    const int t0   = (blk & 63) << 4;
    const int prev = prevArr[b];

    const float* Qb = Q + ((size_t)b * T_ + t0)   * D_;
    const float* Kw = K + ((size_t)b * N_ + prev) * D_;
    const float* Vw = V + ((size_t)b * N_ + prev) * D_;

    // Kick off the V-window DMA before any compute; it hides behind QK+softmax.
    tdm_load_vwin(Vw, (unsigned int)(uintptr_t)&Vlds[0]);

    // Warm the K window stream (global_prefetch_b8).
    __builtin_prefetch(Kw + (size_t)lane * D_, 0, 1);
    __builtin_prefetch(Kw + (size_t)(lane + 32) * D_, 0, 1);

    // ---------------- Phase 1: S = Q_tile @ Kwin^T ----------------
    v8f acc[4] = {};                      // 4 key-tiles of 16; C/D 16x16 f32 frags
    const float* qrow = Qb + (size_t)lrow * D_ + koff;
    const float* krow = Kw + (size_t)lrow * D_ + koff;
    for (int kk = 0; kk < D_; kk += 4) {
        const v2f a = *(const v2f*)(qrow + kk);           // A-frag: Q[row][kk+koff .. +1]
        #pragma unroll
        for (int nt = 0; nt < 4; ++nt) {
            const v2f bk = *(const v2f*)(krow + (size_t)nt * 16 * D_ + kk);
            acc[nt] = __builtin_amdgcn_wmma_f32_16x16x4_f32(
                false, a, false, bk, (short)0, acc[nt], false, false);
        }
    }

    // ---------------- Phase 2: softmax over the 64-wide window ----------------
    const float scale = 0.0625f;          // rsqrt(256)
    #pragma unroll
    for (int r = 0; r < 8; ++r) {
        #pragma unroll
        for (int nt = 0; nt < 4; ++nt) acc[nt][r] *= scale;
        float m = fmaxf(fmaxf(acc[0][r], acc[1][r]), fmaxf(acc[2][r], acc[3][r]));
        m = fmaxf(m, __shfl_xor(m, 8, 32));
        m = fmaxf(m, __shfl_xor(m, 4, 32));
        m = fmaxf(m, __shfl_xor(m, 2, 32));
        m = fmaxf(m, __shfl_xor(m, 1, 32));   // max within each 16-lane half
        float s = 0.f;
        #pragma unroll
        for (int nt = 0; nt < 4; ++nt) { acc[nt][r] = expf(acc[nt][r] - m); s += acc[nt][r]; }
        s += __shfl_xor(s, 8, 32);
        s += __shfl_xor(s, 4, 32);
        s += __shfl_xor(s, 2, 32);
        s += __shfl_xor(s, 1, 32);
        const float inv = 1.0f / s;           // denom: out-of-window terms are exactly 0
        #pragma unroll
        for (int nt = 0; nt < 4; ++nt) acc[nt][r] *= inv;
    }

    // Stage P(16x64) into LDS (C/D layout -> row major)
    #pragma unroll
    for (int nt = 0; nt < 4; ++nt)
        #pragma unroll
        for (int r = 0; r < 8; ++r)
            P[(r + hi8) * 65 + nt * 16 + lrow] = acc[nt][r];
    __syncthreads();

    // argmax per row: first occurrence of max prob == window argmax of reference
    if (lane < 16) {
        float best = P[lane * 65];
        int   bj   = 0;
        for (int j = 1; j < WIN_; ++j) {
            const float v = P[lane * 65 + j];
            if (v > best) { best = v; bj = j; }
        }
        maxatt[(size_t)b * T_ + t0 + lane] = (float)(prev + bj);
    }

    // alignments[b, prev+j, t0+m] = P[m][j]
    {
        float* ab = align_out + (size_t)b * N_ * T_ + (size_t)prev * T_ + t0;
        for (int idx = lane; idx < WIN_ * 16; idx += 32) {
            const int j = idx >> 4, m = idx & 15;
            ab[(size_t)j * T_ + m] = P[m * 65 + j];
        }
    }

    // A-frags of P for all 16 k-chunks
    v2f pa[16];
    #pragma unroll
    for (int c = 0; c < 16; ++c) {
        pa[c].x = P[lrow * 65 + 4 * c + koff];
        pa[c].y = P[lrow * 65 + 4 * c + koff + 1];
    }

    // V window DMA must be complete before the PV phase consumes Vlds.
    __builtin_amdgcn_s_wait_tensorcnt(0);

    // ---------------- Phase 3: O = P @ Vlds, write R = [O | Q] ----------------
    float* Rrow = R + ((size_t)b * T_ + t0) * (2 * D_);
    #pragma unroll 1
    for (int dt = 0; dt < 16; ++dt) {
        v8f o = {};
        const float* vc = &Vlds[(size_t)koff * D_ + dt * 16 + lrow];
        #pragma unroll
        for (int c = 0; c < 16; ++c) {
            v2f vb;
            vb.x = vc[(size_t)(4 * c)     * D_];
            vb.y = vc[(size_t)(4 * c + 1) * D_];
            o = __builtin_amdgcn_wmma_f32_16x16x4_f32(
                false, pa[c], false, vb, (short)0, o, false, false);
        }
        #pragma unroll
        for (int r = 0; r < 8; ++r)
            Rrow[(size_t)(r + hi8) * (2 * D_) + dt * 16 + lrow] = o[r];
    }

    // R[:, D:2D] = Q  (16 rows x 64 float4)
    for (int i = lane; i < 16 * 64; i += 32) {
        const int row = i >> 6, c4 = i & 63;
        ((float4*)(Rrow + (size_t)row * (2 * D_) + D_))[c4] =
            ((const float4*)(Qb + (size_t)row * D_))[c4];
    }
}

// ---------------------------------------------------------------------------
extern "C" void kernel_launch(void* const* d_in, const int* in_sizes, int n_in,
                              void* d_out, int out_size, void* d_ws, size_t ws_size,
                              hipStream_t stream) {
    const float* Q    = (const float*)d_in[0];
    const float* K    = (const float*)d_in[1];
    const float* V    = (const float*)d_in[2];
    const int*   prev = (const int*)d_in[3];

    float* out     = (float*)d_out;
    float* R       = out;                                          // B*T*2D
    float* align_o = out + (size_t)B_ * T_ * 2 * D_;               // B*N*T
    float* maxatt  = align_o + (size_t)B_ * N_ * T_;               // B*T

    // 1) zero alignments (attention kernel overwrites only the 64 window rows)
    zero_align_f4<<<4096, 256, 0, stream>>>((float4*)align_o);

    // 2) windowed attention: one wave32 per (batch, 16-row tile)
    attn_win_wmma<<<B_ * (T_ / 16), 32, 0, stream>>>(Q, K, V, prev,
                                                     R, align_o, maxatt);
}